// GNNEncoder_56573309223083
// MI455X (gfx1250) — compile-verified
//
#include <hip/hip_runtime.h>
#include <hip/hip_bf16.h>

typedef __attribute__((ext_vector_type(2))) float v2f;
typedef __attribute__((ext_vector_type(8))) float v8f;

#define D_EMB 32
#define D_FEAT 16
#define NUM_TYPES 10

// ---------------------------------------------------------------------------
// Zero a float region (grid-stride).
// ---------------------------------------------------------------------------
__global__ void gnn_zero_kernel(float* __restrict__ p, long n) {
    long i = (long)blockIdx.x * blockDim.x + threadIdx.x;
    long stride = (long)gridDim.x * blockDim.x;
    for (; i < n; i += stride) p[i] = 0.0f;
}

// ---------------------------------------------------------------------------
// typed_emb[16][32] = pad16(emb_table[10][16]) @ W[16][32]
// Single wave32. Two 16-wide N tiles; K=16 via 4 chained V_WMMA_F32_16X16X4_F32.
// A layout (16x4 f32): v0 = K {0 | 2} for lane halves, v1 = K {1 | 3}.
// B layout (4x16 f32): v0 = rows {k | k+2} striped over lanes, v1 = {k+1 | k+3}.
// C/D (16x16 f32): VGPR r holds rows {r | r+8} for lane halves.
// ---------------------------------------------------------------------------
__global__ void gnn_wmma_table_kernel(const float* __restrict__ emb,
                                      const float* __restrict__ W,
                                      float* __restrict__ typed) {
    const int lane = threadIdx.x;      // 0..31, all lanes active (EXEC all 1s)
    const int m    = lane & 15;
    const int half = lane >> 4;        // 0: lanes 0-15, 1: lanes 16-31

    for (int tile = 0; tile < 2; ++tile) {
        const int n0 = tile * 16;
        const int n  = n0 + m;
        v8f c = {};
        #pragma unroll
        for (int k = 0; k < 16; k += 4) {
            const int kk = k + half * 2;
            v2f a, b;
            a.x = (m < NUM_TYPES) ? emb[m * D_FEAT + kk]     : 0.0f;
            a.y = (m < NUM_TYPES) ? emb[m * D_FEAT + kk + 1] : 0.0f;
            b.x = W[kk * D_EMB + n];
            b.y = W[(kk + 1) * D_EMB + n];
            c = __builtin_amdgcn_wmma_f32_16x16x4_f32(
                    /*neg_a=*/false, a, /*neg_b=*/false, b,
                    /*c_mod=*/(short)0, c, /*reuse_a=*/false, /*reuse_b=*/false);
        }
        const float* cf = (const float*)&c;
        #pragma unroll
        for (int r = 0; r < 8; ++r) {
            typed[(r + half * 8) * D_EMB + n] = cf[r];
        }
    }
}

// ---------------------------------------------------------------------------
// Degree histograms: one thread per edge, two u32 atomics.
// ---------------------------------------------------------------------------
__global__ void gnn_deg_kernel(const int* __restrict__ src, const int* __restrict__ dst,
                               unsigned* __restrict__ deg_out, unsigned* __restrict__ deg_in,
                               int E) {
    int i = blockIdx.x * blockDim.x + threadIdx.x;
    if (i < E) {
        atomicAdd(&deg_out[src[i]], 1u);
        atomicAdd(&deg_in[dst[i]], 1u);
    }
}

// ---------------------------------------------------------------------------
// In-place deg(u32) -> inv_sqrt(f32). deg==0 -> 0.
// ---------------------------------------------------------------------------
__global__ void gnn_node_prep_kernel(float* __restrict__ inv_out,
                                     float* __restrict__ inv_in, int N) {
    int i = blockIdx.x * blockDim.x + threadIdx.x;
    if (i < N) {
        unsigned d0 = ((const unsigned*)inv_out)[i];
        unsigned d1 = ((const unsigned*)inv_in)[i];
        inv_out[i] = d0 ? rsqrtf((float)d0) : 0.0f;
        inv_in[i]  = d1 ? rsqrtf((float)d1) : 0.0f;
    }
}

// ---------------------------------------------------------------------------
// Edge scatter: one wave per edge, lane = embedding channel.
// agg[dst][c] += typed[type[src]][c] * inv_out[src]   (inv_in factored out)
// typed table (2KB) cached in LDS; per-lane atomic adds are a coalesced
// 128B burst into L2 (agg is 12.8MB -> L2 resident on MI455X's 192MB L2).
// ---------------------------------------------------------------------------
__global__ void gnn_edge_scatter_kernel(const int* __restrict__ src,
                                        const int* __restrict__ dst,
                                        const int* __restrict__ ntype,
                                        const float* __restrict__ inv_out,
                                        const float* __restrict__ typed,
                                        float* __restrict__ agg, int E) {
    __shared__ float styped[D_FEAT * D_EMB];   // 16*32 = 512 floats (rows >=10 are 0)
    for (int i = threadIdx.x; i < D_FEAT * D_EMB; i += blockDim.x) styped[i] = typed[i];
    __syncthreads();

    const int lane = threadIdx.x & 31;
    long warp   = ((long)blockIdx.x * blockDim.x + threadIdx.x) >> 5;
    long nwarps = ((long)gridDim.x * blockDim.x) >> 5;
    for (long e = warp; e < E; e += nwarps) {
        int   s = src[e];            // wave-uniform load (broadcast)
        int   d = dst[e];
        float w = inv_out[s];
        int   t = ntype[s];
        float val = styped[t * D_EMB + lane] * w;   // bank-conflict-free
        atomicAdd(&agg[(size_t)d * D_EMB + lane], val);
    }
}

// ---------------------------------------------------------------------------
// h = relu(agg * inv_in + b); out = mean over nodes, pre-scaled by 1/N.
// Lane = channel, warp strides over nodes; block LDS-reduce -> 32 atomics.
// ---------------------------------------------------------------------------
__global__ void gnn_finalize_kernel(const float* __restrict__ agg,
                                    const float* __restrict__ inv_in,
                                    const float* __restrict__ b,
                                    float* __restrict__ out, int N) {
    const float scale = 1.0f / (float)N;
    const int c  = threadIdx.x & 31;
    const float bc = b[c];

    long node   = ((long)blockIdx.x * blockDim.x + threadIdx.x) >> 5;
    long nwarps = ((long)gridDim.x * blockDim.x) >> 5;
    float acc = 0.0f;
    for (long i = node; i < N; i += nwarps) {
        float v = agg[(size_t)i * D_EMB + c] * inv_in[i] + bc;
        acc += fmaxf(v, 0.0f);
    }

    __shared__ float sm[256];
    sm[threadIdx.x] = acc;
    __syncthreads();
    if (threadIdx.x < 32) {
        float s = 0.0f;
        for (int w = 0; w < (int)(blockDim.x >> 5); ++w) s += sm[w * 32 + threadIdx.x];
        atomicAdd(&out[threadIdx.x], s * scale);
    }
}

// ---------------------------------------------------------------------------
// Launch. Inputs: node_types[N], src[E], dst[E], emb_table[10*16], W[16*32], b[32]
// Output: 32 floats (mean-pooled embedding).
// WS layout (floats): [typed 512][agg N*32][inv_out N][inv_in N]  ~13.6 MB
// ---------------------------------------------------------------------------
extern "C" void kernel_launch(void* const* d_in, const int* in_sizes, int n_in,
                              void* d_out, int out_size, void* d_ws, size_t ws_size,
                              hipStream_t stream) {
    const int*   node_types = (const int*)d_in[0];
    const int*   src        = (const int*)d_in[1];
    const int*   dst        = (const int*)d_in[2];
    const float* emb_table  = (const float*)d_in[3];
    const float* W          = (const float*)d_in[4];
    const float* b          = (const float*)d_in[5];
    float*       out        = (float*)d_out;

    const int N = in_sizes[0];
    const int E = in_sizes[1];

    float* ws      = (float*)d_ws;
    float* typed   = ws;                           // 512 floats
    float* agg     = typed + 512;                  // N*32
    float* inv_out = agg + (size_t)N * D_EMB;      // N (u32 deg, then f32 inv)
    float* inv_in  = inv_out + N;                  // N

    // 1) Zero agg + both degree arrays (contiguous) and the output accumulator.
    long zcount = (long)N * (D_EMB + 2);
    gnn_zero_kernel<<<4096, 256, 0, stream>>>(agg, zcount);
    gnn_zero_kernel<<<1, 32, 0, stream>>>(out, 32);

    // 2) typed_emb = pad(emb_table) @ W via f32 WMMA (single wave).
    gnn_wmma_table_kernel<<<1, 32, 0, stream>>>(emb_table, W, typed);

    // 3) Degree histograms.
    gnn_deg_kernel<<<(E + 255) / 256, 256, 0, stream>>>(
        src, dst, (unsigned*)inv_out, (unsigned*)inv_in, E);

    // 4) deg -> inv_sqrt in place.
    gnn_node_prep_kernel<<<(N + 255) / 256, 256, 0, stream>>>(inv_out, inv_in, N);

    // 5) Edge scatter: wave per edge.
    gnn_edge_scatter_kernel<<<8192, 256, 0, stream>>>(
        src, dst, node_types, inv_out, typed, agg, E);

    // 6) ReLU + mean pool.
    gnn_finalize_kernel<<<256, 256, 0, stream>>>(agg, inv_in, b, out, N);
}